// Yolov2Loss_10084583211513
// MI455X (gfx1250) — compile-verified
//
#include <hip/hip_runtime.h>

typedef __attribute__((ext_vector_type(2))) float v2f;
typedef __attribute__((ext_vector_type(8))) float v8f;

#define YS   52
#define YSS  (YS * YS)        // 2704
#define YA   5
#define YB   128
#define YC   20
#define NCELL (YB * YA * YSS) // 1,730,560
#define NBLK 1040
#define TPB  256

__device__ __forceinline__ float ldnt(const float* p) {
  return __builtin_nontemporal_load(p);
}

// Full-wave (32-lane) f32 sum using V_WMMA_F32_16X16X4_F32.
// A tile: lane l<16 supplies A[l,0]=s, A[l,1]=0; lane l>=16 supplies
// A[l-16,2]=s, A[l-16,3]=0. With B = all-ones, D[M,N] = s_M + s_{M+16}.
// Each lane then holds rows M=0..7 (lanes 0-15) or M=8..15 (lanes 16-31)
// in d[0..7]; summing those + one shfl_xor(16) yields the wave total in f32.
__device__ __forceinline__ float wave_sum32(float s) {
  v2f a; a[0] = s;    a[1] = 0.0f;
  v2f b; b[0] = 1.0f; b[1] = 1.0f;
  v8f c = {};
  v8f d = __builtin_amdgcn_wmma_f32_16x16x4_f32(
      /*neg_a=*/false, a, /*neg_b=*/false, b,
      /*c_mod=*/(short)0, c, /*reuse_a=*/false, /*reuse_b=*/false);
  float r = ((d[0] + d[1]) + (d[2] + d[3])) + ((d[4] + d[5]) + (d[6] + d[7]));
  r += __shfl_xor(r, 16, 32);
  return r;
}

__device__ __forceinline__ float sigmoidf_(float x) {
  return 1.0f / (1.0f + expf(-x));
}

__global__ void yolo_partial_kernel(const float* __restrict__ pred,
                                    const float* __restrict__ targ,
                                    const float* __restrict__ anch,
                                    float* __restrict__ partial) {
  float s_xywh = 0.0f, s_cls = 0.0f, s_obj = 0.0f, s_noobj = 0.0f;

  const int tid    = blockIdx.x * blockDim.x + threadIdx.x;
  const int stride = gridDim.x * blockDim.x;

  for (int t = tid; t < NCELL; t += stride) {
    const int cell = t % YSS;      // i*S + j
    const int ba   = t / YSS;      // b*A + a
    const int a    = ba % YA;
    const int b    = ba / YA;

    // predictions[b, a*25+k, i, j]  -> base + k*YSS (coalesced in j)
    const float* pb = pred + (size_t)25 * (size_t)ba * YSS + cell;
    // targets[b, i, j, a, k]        -> contiguous in k
    const float* tb = targ + ((size_t)(b * YSS + cell) * YA + a) * 25;

    const float conf      = ldnt(pb + 4 * YSS);
    const float obj       = ldnt(tb + 4);
    const float conf_pred = sigmoidf_(conf);

    s_noobj += (1.0f - obj) * conf_pred * conf_pred;

    if (obj != 0.0f) {
      const float tx = ldnt(pb + 0 * YSS);
      const float ty = ldnt(pb + 1 * YSS);
      const float tw = ldnt(pb + 2 * YSS);
      const float th = ldnt(pb + 3 * YSS);
      const float gx = ldnt(tb + 0);
      const float gy = ldnt(tb + 1);
      const float gw = ldnt(tb + 2);
      const float gh = ldnt(tb + 3);
      const float aw = anch[2 * a + 0];
      const float ah = anch[2 * a + 1];

      const float pcx = sigmoidf_(tx);
      const float pcy = sigmoidf_(ty);
      const float pw  = expf(tw) * aw;
      const float ph  = expf(th) * ah;

      const float jf = (float)(cell % YS);
      const float if_ = (float)(cell / YS);

      const float cxg = gx + jf,  cyg = gy + if_;
      const float cxp = pcx + jf, cyp = pcy + if_;

      const float x1min = cxg - gw * 0.5f, x1max = cxg + gw * 0.5f;
      const float y1min = cyg - gh * 0.5f, y1max = cyg + gh * 0.5f;
      const float x2min = cxp - pw * 0.5f, x2max = cxp + pw * 0.5f;
      const float y2min = cyp - ph * 0.5f, y2max = cyp + ph * 0.5f;

      float iw = fminf(x1max, x2max) - fmaxf(x1min, x2min);
      float ih = fminf(y1max, y2max) - fmaxf(y1min, y2min);
      iw = fmaxf(iw, 0.0f);
      ih = fmaxf(ih, 0.0f);
      const float inter = iw * ih;
      const float uni   = gw * gh + pw * ph - inter;
      const float iou   = (uni > 0.0f) ? (inter / uni) : 0.0f;

      const float dx = pcx - gx, dy = pcy - gy;
      const float eps = 1e-6f;
      const float dw = sqrtf(pw + eps) - sqrtf(gw + eps);
      const float dh = sqrtf(ph + eps) - sqrtf(gh + eps);
      s_xywh += obj * ((dx * dx + dy * dy) + (dw * dw + dh * dh));

      const float di = iou - conf_pred;
      s_obj += obj * di * di;

      float sc = 0.0f;
#pragma unroll
      for (int cc = 0; cc < YC; cc++) {
        const float l  = ldnt(pb + (5 + cc) * YSS);
        const float tc = ldnt(tb + 5 + cc);
        // logaddexp(0, l) = max(l,0) + log1p(exp(-|l|))  (stable softplus)
        const float sp = fmaxf(l, 0.0f) + log1pf(expf(-fabsf(l)));
        sc += sp - l * tc;
      }
      s_cls += obj * sc;
    }
  }

  // Wave reduction (WMMA f32, EXEC is all-ones here: 256-thread blocks,
  // loop has uniformly reconverged).
  const float r0 = wave_sum32(s_xywh);
  const float r1 = wave_sum32(s_cls);
  const float r2 = wave_sum32(s_obj);
  const float r3 = wave_sum32(s_noobj);

  __shared__ float wsum[TPB / 32][4];
  const int lane = threadIdx.x & 31;
  const int wid  = threadIdx.x >> 5;
  if (lane == 0) {
    wsum[wid][0] = r0;
    wsum[wid][1] = r1;
    wsum[wid][2] = r2;
    wsum[wid][3] = r3;
  }
  __syncthreads();
  if (threadIdx.x < 4) {
    float s = 0.0f;
#pragma unroll
    for (int w = 0; w < TPB / 32; w++) s += wsum[w][threadIdx.x];
    partial[blockIdx.x * 4 + threadIdx.x] = s;
  }
}

__global__ void yolo_final_kernel(const float* __restrict__ partial,
                                  float* __restrict__ out) {
  __shared__ double red[TPB][4];
  double a0 = 0.0, a1 = 0.0, a2 = 0.0, a3 = 0.0;
  for (int i = threadIdx.x; i < NBLK; i += TPB) {
    const float* p = partial + i * 4;
    a0 += (double)p[0];
    a1 += (double)p[1];
    a2 += (double)p[2];
    a3 += (double)p[3];
  }
  red[threadIdx.x][0] = a0;
  red[threadIdx.x][1] = a1;
  red[threadIdx.x][2] = a2;
  red[threadIdx.x][3] = a3;
  __syncthreads();
  for (int off = TPB / 2; off > 0; off >>= 1) {
    if (threadIdx.x < off) {
#pragma unroll
      for (int c = 0; c < 4; c++)
        red[threadIdx.x][c] += red[threadIdx.x + off][c];
    }
    __syncthreads();
  }
  if (threadIdx.x == 0) {
    const double sxywh  = red[0][0];
    const double scls   = red[0][1];
    const double sobj   = red[0][2];
    const double snoobj = red[0][3];
    out[0] = (float)(5.0 * sxywh + 5.0 * sobj + 0.5 * snoobj + scls);
    out[1] = (float)sxywh;
    out[2] = (float)scls;
    out[3] = (float)(sobj + snoobj);
  }
}

extern "C" void kernel_launch(void* const* d_in, const int* in_sizes, int n_in,
                              void* d_out, int out_size, void* d_ws, size_t ws_size,
                              hipStream_t stream) {
  (void)in_sizes; (void)n_in; (void)out_size; (void)ws_size;
  const float* pred = (const float*)d_in[0];
  const float* targ = (const float*)d_in[1];
  const float* anch = (const float*)d_in[2];
  float* partial = (float*)d_ws;  // NBLK*4 floats, fully overwritten each call
  float* outf    = (float*)d_out;

  hipLaunchKernelGGL(yolo_partial_kernel, dim3(NBLK), dim3(TPB), 0, stream,
                     pred, targ, anch, partial);
  hipLaunchKernelGGL(yolo_final_kernel, dim3(1), dim3(TPB), 0, stream,
                     partial, outf);
}